// KeypointExtract_14474039788263
// MI455X (gfx1250) — compile-verified
//
#include <hip/hip_runtime.h>
#include <hip/hip_bf16.h>
#include <math.h>

// ---- problem constants (from reference) ----
#define BB    4
#define C65   65
#define HC    120
#define WC    160
#define HH    960          // HC*8
#define WW    1280         // WC*8
#define HWI   (HH*WW)      // 1,228,800
#define NCAND 8192
#define NKPT  1024
#define PAD   4
#define BORD  4
#define CONF  0.015f

// padded NMS bit-grid: (HH+2*PAD) rows x (WW+2*PAD)=1288 bits -> 41 words/row
#define GROWS (HH + 2*PAD)          // 968
#define GWORDS 41                   // ceil(1288/32)

typedef float v2f __attribute__((ext_vector_type(2)));
typedef float v8f __attribute__((ext_vector_type(8)));

// ---------------------------------------------------------------------------
// K1: channel softmax over 65 + drop dust + depth2space -> heatmap (f32 out)
// ---------------------------------------------------------------------------
__global__ void k_softmax_d2s(const float* __restrict__ semi,
                              float* __restrict__ heat) {
    int t = blockIdx.x * blockDim.x + threadIdx.x;
    if (t >= BB * HC * WC) return;
    int wc = t % WC;
    int hc = (t / WC) % HC;
    int b  = t / (WC * HC);
    const float* base = semi + ((size_t)b * C65 * HC * WC) + (size_t)hc * WC + wc;

    float m = -3.4e38f;
    for (int ch = 0; ch < C65; ++ch)
        m = fmaxf(m, base[(size_t)ch * HC * WC]);
    float s = 0.f;
    for (int ch = 0; ch < C65; ++ch)
        s += expf(base[(size_t)ch * HC * WC] - m);
    float inv = 1.f / s;

    float* hb = heat + (size_t)b * HWI;
    for (int ch = 0; ch < 64; ++ch) {
        float e = expf(base[(size_t)ch * HC * WC] - m) * inv;
        int r = ch >> 3, c = ch & 7;
        hb[(size_t)(hc * 8 + r) * WW + (wc * 8 + c)] = e;
    }
}

// ---------------------------------------------------------------------------
// K2: exact k-th largest (k=8192) per batch via 4-pass MSD radix select on
// the positive-float bit pattern. Also zeroes the compaction counters.
// ---------------------------------------------------------------------------
__global__ void __launch_bounds__(1024)
k_radix_select(const float* __restrict__ heat,
               unsigned* __restrict__ thresh, unsigned* __restrict__ cnt) {
    int b = blockIdx.x;
    const float* hb = heat + (size_t)b * HWI;
    __shared__ unsigned hist[256];
    __shared__ unsigned sh_prefix, sh_k;
    if (threadIdx.x == 0) { sh_prefix = 0u; sh_k = NCAND; }
    __syncthreads();
    for (int pass = 0; pass < 4; ++pass) {
        int s = 24 - 8 * pass;
        for (int i = threadIdx.x; i < 256; i += blockDim.x) hist[i] = 0u;
        __syncthreads();
        unsigned prefix = sh_prefix;
        for (int i = threadIdx.x; i < HWI; i += blockDim.x) {
            unsigned key = __float_as_uint(hb[i]);
            bool match = (pass == 0) || ((key >> (s + 8)) == (prefix >> (s + 8)));
            if (match) atomicAdd(&hist[(key >> s) & 255u], 1u);
        }
        __syncthreads();
        if (threadIdx.x == 0) {
            unsigned k = sh_k, cum = 0u; int d = 0;
            for (d = 255; d >= 0; --d) { cum += hist[d]; if (cum >= k) break; }
            if (d < 0) d = 0;
            sh_k = k - (cum - hist[d]);
            sh_prefix = prefix | ((unsigned)d << s);
        }
        __syncthreads();
    }
    if (threadIdx.x == 0) { thresh[b] = sh_prefix; cnt[b] = 0u; }
}

// ---------------------------------------------------------------------------
// K3: compact all elements with key >= thresh into composite-key candidates
// composite = (valbits << 32) | ~pixelIndex  (descending sort => value desc,
// index ascending on ties, matching jax top_k)
// ---------------------------------------------------------------------------
__global__ void k_compact(const float* __restrict__ heat,
                          const unsigned* __restrict__ thresh,
                          unsigned* __restrict__ cnt,
                          unsigned long long* __restrict__ cand) {
    size_t t = (size_t)blockIdx.x * blockDim.x + threadIdx.x;
    if (t >= (size_t)BB * HWI) return;
    int b = (int)(t / HWI);
    unsigned i = (unsigned)(t % HWI);
    unsigned key = __float_as_uint(heat[t]);
    if (key >= thresh[b]) {
        unsigned slot = atomicAdd(&cnt[b], 1u);
        if (slot < NCAND)
            cand[(size_t)b * NCAND + slot] =
                ((unsigned long long)key << 32) | (unsigned)(~i);
    }
}

// ---------------------------------------------------------------------------
// K4: bitonic sort 8192 composite keys descending in 64KB LDS, per batch.
// ---------------------------------------------------------------------------
__global__ void __launch_bounds__(1024)
k_sort(const unsigned long long* __restrict__ cand,
       const unsigned* __restrict__ cnt,
       float* __restrict__ svals, int* __restrict__ sidx) {
    int b = blockIdx.x;
    __shared__ unsigned long long sk[NCAND];
    unsigned n = cnt[b]; if (n > NCAND) n = NCAND;
    for (int j = threadIdx.x; j < NCAND; j += blockDim.x)
        sk[j] = (j < (int)n) ? cand[(size_t)b * NCAND + j] : 0ull;
    __syncthreads();
    for (int k = 2; k <= NCAND; k <<= 1) {
        for (int j = k >> 1; j > 0; j >>= 1) {
            for (int t = threadIdx.x; t < NCAND; t += blockDim.x) {
                int l = t ^ j;
                if (l > t) {
                    unsigned long long a = sk[t], c = sk[l];
                    bool descBlk = ((t & k) == 0);
                    bool doSwap = descBlk ? (a < c) : (a > c);
                    if (doSwap) { sk[t] = c; sk[l] = a; }
                }
            }
            __syncthreads();
        }
    }
    for (int j = threadIdx.x; j < NCAND; j += blockDim.x) {
        unsigned long long v = sk[j];
        svals[(size_t)b * NCAND + j] = __uint_as_float((unsigned)(v >> 32));
        sidx[(size_t)b * NCAND + j]  = (int)(~(unsigned)(v & 0xFFFFFFFFull));
    }
}

// ---------------------------------------------------------------------------
// K5: sequential greedy NMS. Padded bit-grid (968 x 41 words = ~155KB) lives
// entirely in CDNA5's 320KB WGP LDS. One wave per batch; lane 0 tests the
// center bit, 18 lanes OR the 9x9 patch (each row spans <=2 words, all
// distinct). Same-wave DS ops are in-order so no per-iteration barrier is
// required. Then lane 0 serially takes the first 1024 kept&border candidates
// (filling with earliest non-selected slots if short, matching top_k -inf
// tie-breaking).
// ---------------------------------------------------------------------------
__global__ void k_nms(const float* __restrict__ svals,
                      const int* __restrict__ sidx,
                      int* __restrict__ selpos) {
    int b = blockIdx.x;
    __shared__ unsigned nmsgrid[GROWS * GWORDS];
    __shared__ unsigned keptb[NCAND / 32];
    volatile unsigned* g = nmsgrid;
    int lane = threadIdx.x;
    for (int i = lane; i < GROWS * GWORDS; i += 32) nmsgrid[i] = 0u;
    for (int i = lane; i < NCAND / 32; i += 32) keptb[i] = 0u;
    __syncthreads();

    const float* vb = svals + (size_t)b * NCAND;
    const int*   ib = sidx  + (size_t)b * NCAND;

    for (int j = 0; j < NCAND; ++j) {
        int keep = 0, y = 0, x = 0;
        if (lane == 0) {
            float v = vb[j];
            int p = ib[j];
            if (p >= 0 && p < HWI && v >= CONF) {
                y = p / WW; x = p - y * WW;
                unsigned cw = g[(y + PAD) * GWORDS + ((x + PAD) >> 5)];
                keep = ((cw >> ((x + PAD) & 31)) & 1u) ? 0 : 1;
            }
        }
        keep = __shfl(keep, 0, 32);
        y    = __shfl(y, 0, 32);
        x    = __shfl(x, 0, 32);
        if (keep) {
            if (lane < 18) {
                int r = lane >> 1, w = lane & 1;
                int fw = x >> 5, sh = x & 31;
                unsigned long long mask = 0x1FFull << sh;
                unsigned mw = w ? (unsigned)(mask >> 32) : (unsigned)(mask & 0xFFFFFFFFull);
                if (mw) {
                    int widx = (y + r) * GWORDS + fw + w;
                    g[widx] = g[widx] | mw;
                }
            }
            if (lane == 0) keptb[j >> 5] |= (1u << (j & 31));
        }
    }
    __syncthreads();

    if (lane == 0) {
        int c = 0;
        int* sp = selpos + b * NKPT;
        for (int j = 0; j < NCAND && c < NKPT; ++j) {
            if ((keptb[j >> 5] >> (j & 31)) & 1u) {
                int p = ib[j]; int y = p / WW, x = p - y * WW;
                if (x >= BORD && x < WW - BORD && y >= BORD && y < HH - BORD)
                    sp[c++] = p;
            }
        }
        if (c < NKPT) {
            for (int j = 0; j < NCAND && c < NKPT; ++j) {
                int p = ib[j]; int y = p / WW, x = p - y * WW;
                bool kb = (((keptb[j >> 5] >> (j & 31)) & 1u) != 0u) &&
                          x >= BORD && x < WW - BORD && y >= BORD && y < HH - BORD;
                if (!kb) sp[c++] = p;
            }
        }
    }
}

// ---------------------------------------------------------------------------
// K6: 5x5 soft-argmax refinement via WMMA. One wave handles 16 keypoints.
// Out(3x16) = Coef(3x25, padded to 16x25) x E(25x16), accumulated as seven
// fully-unrolled V_WMMA_F32_16X16X4_F32 (K=4 chunks; columns 25..27 of A are
// identically zero, so garbage-but-finite E taps there contribute nothing).
//   A layout (16x4 f32): lanes 0-15 K={0,1} in v.x/v.y, lanes 16-31 K={2,3}
//   B layout (4x16 f32): lane (l&15)=N, same K split.
//   D layout: lane l<16 holds N=l, acc[0..2] = rows 0..2 = {S, Sy, Sx}.
// Reference offsets: region r = i*5+j, off0(y)=j-2, off1(x)=i-2.
// Pipeline: (1) issue all 14 tap loads, (2) exps, (3) 7 back-to-back WMMAs.
// Entirely branchless (selects only) -> no exec-mask divergence around WMMA.
// ---------------------------------------------------------------------------
__global__ void k_refine(const float* __restrict__ heat,
                         const int* __restrict__ selpos,
                         const float* __restrict__ temp,
                         float* __restrict__ outPts) {
    int bid = blockIdx.x;          // BB*64 blocks, 32 threads each
    int b = bid >> 6;
    int grp = bid & 63;
    int l = threadIdx.x;
    int nloc = l & 15;
    int n = grp * 16 + nloc;
    int p = selpos[b * NKPT + n];
    int y = p / WW, x = p - y * WW;
    float T = temp[0];
    const float* hb = heat + (size_t)b * HWI;
    int m = nloc;                  // coefficient row for A operand
    int khalf = (l >> 4) << 1;     // 0 for lanes 0-15, 2 for lanes 16-31

    float hv[14], av[14];
#pragma unroll
    for (int c = 0; c < 7; ++c) {
#pragma unroll
        for (int h = 0; h < 2; ++h) {
            int idx = 2 * c + h;
            int r = 4 * c + khalf + h;            // 0..27 (per-lane)
            int rq = r / 5;                       // tap column i
            int oy = (r - rq * 5) - 2;            // off0 = lin[j]
            int ox = rq - 2;                      // off1 = lin[i]
            int yy = y + oy; yy = (yy < 0) ? yy + HH : yy;
            yy = min(max(yy, 0), HH - 1);
            int xx = x + ox; xx = (xx < 0) ? xx + WW : xx;
            xx = min(max(xx, 0), WW - 1);
            hv[idx] = hb[(size_t)yy * WW + xx];
            float valid = (r < 25) ? 1.f : 0.f;
            float coef = (m == 0) ? 1.f
                       : (m == 1) ? (float)oy
                       : (m == 2) ? (float)ox : 0.f;
            av[idx] = valid * coef;               // zero-pad columns r>=25
        }
    }

    float ev[14];
#pragma unroll
    for (int i = 0; i < 14; ++i)
        ev[i] = __expf(hv[i] * T);                // finite (heat<=1, T=10)

    v8f acc = {0.f, 0.f, 0.f, 0.f, 0.f, 0.f, 0.f, 0.f};
#pragma unroll
    for (int c = 0; c < 7; ++c) {
        v2f a2 = {av[2 * c], av[2 * c + 1]};
        v2f e2 = {ev[2 * c], ev[2 * c + 1]};
        acc = __builtin_amdgcn_wmma_f32_16x16x4_f32(
            /*neg_a=*/false, a2, /*neg_b=*/false, e2,
            /*c_mod=*/(short)0, acc, /*reuse_a=*/false, /*reuse_b=*/false);
    }

    if (l < 16) {
        float S = acc[0], Sy = acc[1], Sx = acc[2];
        float dy = Sy / S, dx = Sx / S;
        float xf = (float)x + dx;
        float yf = (float)y + dy;
        int base = (b * NKPT + n) * 2;
        outPts[base + 0] = xf;     // batch_pts = [x, y]
        outPts[base + 1] = yf;
    }
}

// ---------------------------------------------------------------------------
// K7: bilinear grid-sample of coarse_desc (256 ch) + L2 normalize.
// One block per keypoint, thread = channel; LDS tree-reduce for the norm.
// ix = x_f/8 - 0.5 (since W/Wc = 8). Zero-pad out-of-bounds taps.
// ---------------------------------------------------------------------------
__global__ void k_desc(const float* __restrict__ cdesc,
                       const float* __restrict__ pts,
                       float* __restrict__ desc) {
    int kp = blockIdx.x;           // BB*NKPT
    int b = kp >> 10;
    int n = kp & 1023;
    int c = threadIdx.x;           // 0..255
    float xf = pts[kp * 2 + 0];
    float yf = pts[kp * 2 + 1];
    float ix = xf * 0.125f - 0.5f;
    float iy = yf * 0.125f - 0.5f;
    float x0f = floorf(ix), y0f = floorf(iy);
    int x0 = (int)x0f, y0 = (int)y0f;
    float wx1 = ix - x0f, wy1 = iy - y0f;
    float w00 = (1.f - wx1) * (1.f - wy1);
    float w01 = wx1 * (1.f - wy1);
    float w10 = (1.f - wx1) * wy1;
    float w11 = wx1 * wy1;
    const float* cb = cdesc + ((size_t)b * 256 + c) * (HC * WC);

    float v = 0.f;
    {
        int xi, yi;
        xi = x0;     yi = y0;
        if (xi >= 0 && xi < WC && yi >= 0 && yi < HC) v += w00 * cb[yi * WC + xi];
        xi = x0 + 1; yi = y0;
        if (xi >= 0 && xi < WC && yi >= 0 && yi < HC) v += w01 * cb[yi * WC + xi];
        xi = x0;     yi = y0 + 1;
        if (xi >= 0 && xi < WC && yi >= 0 && yi < HC) v += w10 * cb[yi * WC + xi];
        xi = x0 + 1; yi = y0 + 1;
        if (xi >= 0 && xi < WC && yi >= 0 && yi < HC) v += w11 * cb[yi * WC + xi];
    }

    __shared__ float red[256];
    red[c] = v * v;
    __syncthreads();
    for (int s = 128; s > 0; s >>= 1) {
        if (c < s) red[c] += red[c + s];
        __syncthreads();
    }
    float norm = sqrtf(red[0]);
    float scale = 1.f / fmaxf(norm, 1e-12f);
    desc[((size_t)b * 256 + c) * NKPT + n] = v * scale;
}

// ---------------------------------------------------------------------------
// launcher
// ---------------------------------------------------------------------------
extern "C" void kernel_launch(void* const* d_in, const int* in_sizes, int n_in,
                              void* d_out, int out_size, void* d_ws, size_t ws_size,
                              hipStream_t stream) {
    (void)in_sizes; (void)n_in; (void)out_size; (void)ws_size;
    const float* semi  = (const float*)d_in[0];   // (4,65,120,160)
    const float* cdesc = (const float*)d_in[1];   // (4,256,120,160)
    const float* temp  = (const float*)d_in[2];   // (1,)

    float* out = (float*)d_out;
    float* outPts  = out;                                   // (4,1024,2)
    float* outDesc = out + (size_t)BB * NKPT * 2;           // (4,256,1024)
    float* outHeat = outDesc + (size_t)BB * 256 * NKPT;     // (4,960,1280)

    char* ws = (char*)d_ws;
    unsigned* thresh            = (unsigned*)(ws + 0);        // 16 B
    unsigned* cnt               = (unsigned*)(ws + 16);       // 16 B
    unsigned long long* cand    = (unsigned long long*)(ws + 64);        // 4*8192*8
    float* svals                = (float*)(ws + 64 + (size_t)BB*NCAND*8);            // 4*8192*4
    int*   sidx                 = (int*)  (ws + 64 + (size_t)BB*NCAND*8 + (size_t)BB*NCAND*4);
    int*   selpos               = (int*)  (ws + 64 + (size_t)BB*NCAND*8 + 2*(size_t)BB*NCAND*4);

    k_softmax_d2s<<<(BB * HC * WC + 255) / 256, 256, 0, stream>>>(semi, outHeat);
    k_radix_select<<<BB, 1024, 0, stream>>>(outHeat, thresh, cnt);
    k_compact<<<(int)(((size_t)BB * HWI + 255) / 256), 256, 0, stream>>>(outHeat, thresh, cnt, cand);
    k_sort<<<BB, 1024, 0, stream>>>(cand, cnt, svals, sidx);
    k_nms<<<BB, 32, 0, stream>>>(svals, sidx, selpos);
    k_refine<<<BB * 64, 32, 0, stream>>>(outHeat, selpos, temp, outPts);
    k_desc<<<BB * NKPT, 256, 0, stream>>>(cdesc, outPts, outDesc);
}